// FusedMHC_29721173689133
// MI455X (gfx1250) — compile-verified
//
#include <hip/hip_runtime.h>

typedef __attribute__((ext_vector_type(8))) float v8f;
typedef __attribute__((ext_vector_type(2))) float v2f;

// 72 % 64 == 8 -> matrices start 8 banks apart; column reads (stride 8 floats
// within a matrix, +8 banks per matrix) touch 32 distinct banks per wave.
#define MAT_STRIDE 72

// One wave32 owns 4 independent 8x8 matrices (1KB), fully in registers.
// Sinkhorn via cumulative scaling vectors:  out = R_i * exp(x)_ij * C_j with
//   R = 1/(M C_prev)   (row step),   C = 1/(R^T M)   (col step).
// Both mat-vec reductions run on the XDL pipe: data (weighted by the scale
// vector) is the WMMA B operand, a constant 0/1 block-selection matrix is A,
// so V_WMMA_F32_16X16X4_F32 sums each lane's 8 registers and broadcasts the
// result into every D register of that lane.
__global__ __launch_bounds__(256) void sinkhorn8x8_kernel(
    const float* __restrict__ x, const int* __restrict__ nit,
    float* __restrict__ out, int numTiles) {
  __shared__ float lds[8 * 4 * MAT_STRIDE];   // 8 waves * 4 matrices * 72 floats

  const int tid  = blockIdx.x * blockDim.x + threadIdx.x;
  const int tile = tid >> 5;                  // one wave per 4-matrix tile
  if (tile >= numTiles) return;               // wave-uniform: EXEC stays all-ones
  const int lane  = threadIdx.x & 31;
  const int iters = *nit;

  const int mat = lane >> 3;                  // matrix within tile (0..3)
  const int row = lane & 7;                   // lane's row (orientation 0)

  // ---- load: lane = one row of one matrix (8 consecutive floats) ----
  const long base = (long)tile * 256 + (long)(mat * 64 + row * 8);
  const float4 lo = *(const float4*)(x + base);
  const float4 hi = *(const float4*)(x + base + 4);
  float m0[8] = {lo.x, lo.y, lo.z, lo.w, hi.x, hi.y, hi.z, hi.w};
  #pragma unroll
  for (int i = 0; i < 8; ++i)
    m0[i] = __builtin_amdgcn_exp2f(m0[i] * 1.44269504088896340736f);

  // ---- one-time LDS transpose -> u[v] = M[v][row]  (lane = column) ----
  // Wave-private region; same-wave LDS ops execute in order, no barrier needed.
  float* myLds = lds + (threadIdx.x >> 5) * (4 * MAT_STRIDE) + mat * MAT_STRIDE;
  float4 s0; s0.x = m0[0]; s0.y = m0[1]; s0.z = m0[2]; s0.w = m0[3];
  float4 s1; s1.x = m0[4]; s1.y = m0[5]; s1.z = m0[6]; s1.w = m0[7];
  *(float4*)(myLds + row * 8)     = s0;
  *(float4*)(myLds + row * 8 + 4) = s1;
  float u[8];
  #pragma unroll
  for (int v = 0; v < 8; ++v) u[v] = myLds[v * 8 + row];   // conflict-free

  // ---- constant WMMA A operand: A[m][k] = 1 iff (k<2)==(m<8) ----
  // A-layout: m = lane%16, k = v + 2*(lane/16); with v in {0,1} the value is
  // lane-uniform across both A VGPRs.
  const float asel = (((lane & 15) < 8) == (lane < 16)) ? 1.0f : 0.0f;
  v2f A; A.x = asel; A.y = asel;

  // ---- absolute-index allgather indices (byte addresses for ds_bpermute) ----
  int bidx[8];
  #pragma unroll
  for (int j = 0; j < 8; ++j) bidx[j] = ((lane & 24) + j) << 2;

  float c8[8];
  #pragma unroll
  for (int j = 0; j < 8; ++j) c8[j] = 1.0f;
  float r = 1.0f;

  for (int it = 0; it < iters; ++it) {
    // s = M * C : weight columns by C, WMMA sums each lane's row -> replicated
    v8f acc = {0.f, 0.f, 0.f, 0.f, 0.f, 0.f, 0.f, 0.f};
    #pragma unroll
    for (int p = 0; p < 8; p += 2) {
      v2f B; B.x = m0[p] * c8[p]; B.y = m0[p + 1] * c8[p + 1];
      acc = __builtin_amdgcn_wmma_f32_16x16x4_f32(false, A, false, B,
                                                  (short)0, acc, false, false);
    }
    r = __builtin_amdgcn_rcpf(acc[0]);        // lane's cumulative row scale

    // allgather R across the 8 lanes of this matrix (absolute order)
    float r8[8];
    #pragma unroll
    for (int j = 0; j < 8; ++j)
      r8[j] = __int_as_float(
          __builtin_amdgcn_ds_bpermute(bidx[j], __float_as_int(r)));

    // t = R^T * M : weight rows by R on the transposed copy, WMMA-sum in-lane
    v8f acc2 = {0.f, 0.f, 0.f, 0.f, 0.f, 0.f, 0.f, 0.f};
    #pragma unroll
    for (int p = 0; p < 8; p += 2) {
      v2f B; B.x = u[p] * r8[p]; B.y = u[p + 1] * r8[p + 1];
      acc2 = __builtin_amdgcn_wmma_f32_16x16x4_f32(false, A, false, B,
                                                   (short)0, acc2, false, false);
    }
    const float c = __builtin_amdgcn_rcpf(acc2[0]);  // lane's col scale

    // allgather C (absolute order) for the next iteration / final scaling
    #pragma unroll
    for (int j = 0; j < 8; ++j)
      c8[j] = __int_as_float(
          __builtin_amdgcn_ds_bpermute(bidx[j], __float_as_int(c)));
  }

  // ---- out = R_i * M_ij * C_j ----
  float o[8];
  #pragma unroll
  for (int j = 0; j < 8; ++j) o[j] = r * m0[j] * c8[j];
  float4 olo; olo.x = o[0]; olo.y = o[1]; olo.z = o[2]; olo.w = o[3];
  float4 ohi; ohi.x = o[4]; ohi.y = o[5]; ohi.z = o[6]; ohi.w = o[7];
  *(float4*)(out + base)     = olo;
  *(float4*)(out + base + 4) = ohi;
}

extern "C" void kernel_launch(void* const* d_in, const int* in_sizes, int n_in,
                              void* d_out, int out_size, void* d_ws, size_t ws_size,
                              hipStream_t stream) {
  const float* x   = (const float*)d_in[0];
  const int*   nit = (const int*)d_in[1];
  float*       outp = (float*)d_out;

  const int total    = in_sizes[0];     // B * 64 floats
  const int numTiles = total / 256;     // 4 matrices (256 floats) per wave
  const long threads = (long)numTiles * 32;
  const int  block   = 256;             // 8 waves per block
  const int  grid    = (int)((threads + block - 1) / block);

  sinkhorn8x8_kernel<<<grid, block, 0, stream>>>(x, nit, outp, numTiles);
}